// VectorQuantizer_13305808683335
// MI455X (gfx1250) — compile-verified
//
#include <hip/hip_runtime.h>
#include <hip/hip_bf16.h>

typedef __attribute__((ext_vector_type(2))) float v2f;
typedef __attribute__((ext_vector_type(8))) float v8f;

#define K_EMB   512
#define C_DIM   64
#define SPAT    32768              // 32*32*32
#define NBATCH  4
#define TILE_S  256                // rows per workgroup
#define THREADS 256                // 8 waves (wave32)
#define RG      2                  // 16-row groups per wave (32 rows/wave)
#define PCB_STRIDE 528             // float2 stride per kp-row: 512 + 16 pad -> conflict-free halves
#define LDS_PCB_BYTES (32 * PCB_STRIDE * 8)            // 135168
#define LDS_E2_BYTES  (K_EMB * 4)                      // 2048
#define LDS_IDX_BYTES (8 * RG * 16 * 4)                // 1024
#define LDS_WSUM_BYTES (8 * 4)
#define LDS_TOTAL (LDS_PCB_BYTES + LDS_E2_BYTES + LDS_IDX_BYTES + LDS_WSUM_BYTES)

__global__ __launch_bounds__(THREADS) void vq_main(
    const float* __restrict__ x,     // [4][64][32768]
    const float* __restrict__ cb,    // [512][64]
    float* __restrict__ outq,        // [4][64][32768]
    float* __restrict__ partial)     // [gridDim.x]
{
    extern __shared__ char smem[];
    float2* pcb  = reinterpret_cast<float2*>(smem);                                   // [32][528] pairs (cb[n][2kp],cb[n][2kp+1])
    float*  e2s  = reinterpret_cast<float*>(smem + LDS_PCB_BYTES);                    // [512]
    int*    idxl = reinterpret_cast<int*>(smem + LDS_PCB_BYTES + LDS_E2_BYTES);       // [8*32]
    float*  wsum = reinterpret_cast<float*>(smem + LDS_PCB_BYTES + LDS_E2_BYTES + LDS_IDX_BYTES);

    const int tid  = threadIdx.x;
    const int lane = tid & 31;
    const int wave = tid >> 5;
    const int half = lane >> 4;     // which 16-lane half of the wave
    const int l16  = lane & 15;

    // ---- Phase 1: stream codebook rows -> transposed pair layout in LDS + ||e||^2
    for (int n = tid; n < K_EMB; n += THREADS) {
        float4 r[16];
        const float4* row = reinterpret_cast<const float4*>(cb + n * C_DIM);
        #pragma unroll
        for (int q = 0; q < 16; ++q) r[q] = row[q];          // 256B contiguous per lane
        const float* rf = reinterpret_cast<const float*>(r);
        float s2 = 0.f;
        #pragma unroll
        for (int q = 0; q < C_DIM; ++q) s2 += rf[q] * rf[q];
        #pragma unroll
        for (int kp = 0; kp < 32; ++kp) {                    // b64 LDS stores, bank-conflict-free
            float2 p; p.x = rf[2 * kp]; p.y = rf[2 * kp + 1];
            pcb[kp * PCB_STRIDE + n] = p;
        }
        e2s[n] = s2;
    }
    __syncthreads();

    const int b  = blockIdx.x >> 7;            // 128 tiles per batch
    const int s0 = (blockIdx.x & 127) * TILE_S;
    const int sW = s0 + wave * (TILE_S / 8);   // 32 rows per wave
    const float* xb = x + (size_t)b * C_DIM * SPAT;

    // ---- Phase 2: A fragments in registers.
    // WMMA f32 16x16x4 A layout: lane l16 = M; VGPR0 = {K0 | K2}, VGPR1 = {K1 | K3}.
    v2f a[RG][16];
    #pragma unroll
    for (int rg = 0; rg < RG; ++rg) {
        const int row = sW + rg * 16 + l16;
        #pragma unroll
        for (int kk = 0; kk < 16; ++kk) {
            const int k0 = 4 * kk + 2 * half;
            a[rg][kk].x = __builtin_nontemporal_load(&xb[(size_t)k0 * SPAT + row]);
            a[rg][kk].y = __builtin_nontemporal_load(&xb[(size_t)(k0 + 1) * SPAT + row]);
        }
    }

    float bd[RG][8]; int bi[RG][8];
    #pragma unroll
    for (int rg = 0; rg < RG; ++rg)
        #pragma unroll
        for (int i = 0; i < 8; ++i) { bd[rg][i] = 3.4e38f; bi[rg][i] = 0; }

    // ---- Phase 3: sweep 32 codebook column tiles; D = A x B via v_wmma_f32_16x16x4_f32
    for (int ct = 0; ct < K_EMB / 16; ++ct) {
        const int n = ct * 16 + l16;           // this lane's codebook column
        v2f bb[16];
        #pragma unroll
        for (int kk = 0; kk < 16; ++kk)        // B layout mirrors A: {K0|K2},{K1|K3}, N = l16
            bb[kk] = *reinterpret_cast<const v2f*>(&pcb[(2 * kk + half) * PCB_STRIDE + n]);

        v8f acc0 = {0.f,0.f,0.f,0.f,0.f,0.f,0.f,0.f};
        v8f acc1 = {0.f,0.f,0.f,0.f,0.f,0.f,0.f,0.f};
        #pragma unroll
        for (int kk = 0; kk < 16; ++kk) {      // two independent accumulation chains
            acc0 = __builtin_amdgcn_wmma_f32_16x16x4_f32(false, a[0][kk], false, bb[kk],
                                                         (short)0, acc0, false, false);
            acc1 = __builtin_amdgcn_wmma_f32_16x16x4_f32(false, a[1][kk], false, bb[kk],
                                                         (short)0, acc1, false, false);
        }
        const float e2v = e2s[n];
        #pragma unroll
        for (int i = 0; i < 8; ++i) {          // dist = ||e||^2 - 2 x.e  (||x||^2 constant per row)
            float d0 = e2v - 2.f * acc0[i];
            if (d0 < bd[0][i]) { bd[0][i] = d0; bi[0][i] = n; }
            float d1 = e2v - 2.f * acc1[i];
            if (d1 < bd[1][i]) { bd[1][i] = d1; bi[1][i] = n; }
        }
    }

    // ---- Phase 4: argmin reduction across the 16 lanes holding each row (first-min tie-break)
    #pragma unroll
    for (int rg = 0; rg < RG; ++rg)
        #pragma unroll
        for (int i = 0; i < 8; ++i) {
            float d = bd[rg][i]; int ix = bi[rg][i];
            #pragma unroll
            for (int off = 1; off <= 8; off <<= 1) {
                float d2 = __shfl_xor(d, off, 32);
                int   x2 = __shfl_xor(ix, off, 32);
                if (d2 < d || (d2 == d && x2 < ix)) { d = d2; ix = x2; }
            }
            bi[rg][i] = ix;
        }
    if (l16 == 0) {                             // C-layout: VGPR i -> row i + 8*half
        #pragma unroll
        for (int rg = 0; rg < RG; ++rg)
            #pragma unroll
            for (int i = 0; i < 8; ++i)
                idxl[wave * 32 + rg * 16 + half * 8 + i] = bi[rg][i];
    }
    __syncthreads();

    // ---- Phase 5a: loss from resident A registers (each row split across lane L and L+16)
    float ls = 0.f;
    #pragma unroll
    for (int rg = 0; rg < RG; ++rg) {
        const int ownIdx = idxl[wave * 32 + rg * 16 + l16];
        #pragma unroll
        for (int kk = 0; kk < 16; ++kk) {
            float2 cv = pcb[(2 * kk + half) * PCB_STRIDE + ownIdx];
            float dx = cv.x - a[rg][kk].x;
            float dy = cv.y - a[rg][kk].y;
            ls += dx * dx + dy * dy;
        }
    }
    #pragma unroll
    for (int off = 16; off >= 1; off >>= 1) ls += __shfl_xor(ls, off, 32);
    if (lane == 0) wsum[wave] = ls;

    // ---- Phase 5b: gather + store quantized output, full-wave contiguous 128B per channel
    const int myRow = sW + lane;                // 32 rows per wave, one per lane
    const int myIdx = idxl[wave * 32 + lane];
    #pragma unroll
    for (int c2 = 0; c2 < 32; ++c2) {
        float2 cv = pcb[c2 * PCB_STRIDE + myIdx];
        __builtin_nontemporal_store(cv.x, &outq[((size_t)b * C_DIM + 2 * c2) * SPAT + myRow]);
        __builtin_nontemporal_store(cv.y, &outq[((size_t)b * C_DIM + 2 * c2 + 1) * SPAT + myRow]);
    }

    __syncthreads();
    if (tid == 0) {
        float t = 0.f;
        #pragma unroll
        for (int w = 0; w < 8; ++w) t += wsum[w];
        partial[blockIdx.x] = t;                // fixed order -> deterministic
    }
}

__global__ void vq_finish(const float* __restrict__ partial, float* __restrict__ loss)
{
    const int lane = threadIdx.x;               // single wave32
    float s = 0.f;
    for (int i = lane; i < 512; i += 32) s += partial[i];
    #pragma unroll
    for (int off = 16; off >= 1; off >>= 1) s += __shfl_xor(s, off, 32);
    if (lane == 0)
        loss[0] = 1.25f * s * (1.0f / 8388608.0f);  // (q_loss + 0.25*e_loss), identical means
}

extern "C" void kernel_launch(void* const* d_in, const int* in_sizes, int n_in,
                              void* d_out, int out_size, void* d_ws, size_t ws_size,
                              hipStream_t stream)
{
    const float* x  = (const float*)d_in[0];   // [4,64,32,32,32] fp32
    const float* cb = (const float*)d_in[1];   // [512,64] fp32
    float* loss = (float*)d_out;               // scalar first
    float* outq = (float*)d_out + 1;           // then [4,64,32,32,32]
    float* partial = (float*)d_ws;             // 512 floats scratch

    vq_main<<<dim3(512), dim3(THREADS), LDS_TOTAL, stream>>>(x, cb, outq, partial);
    vq_finish<<<dim3(1), dim3(32), 0, stream>>>(partial, loss);
}